// EncoderLayer_13151189860787
// MI455X (gfx1250) — compile-verified
//
#include <hip/hip_runtime.h>
#include <math.h>

typedef __attribute__((ext_vector_type(2))) float v2f;
typedef __attribute__((ext_vector_type(4))) float v4f;
typedef __attribute__((ext_vector_type(8))) float v8f;

#define SW 132              // LDS row stride (floats): conflict-free for A/B frag patterns
#define LDS_FLOATS 67584    // 4 * 128*132 = 270,336 bytes (of 320KB/WGP)

static __device__ __forceinline__ v8f wmma4(v2f a, v2f b, v8f c) {
  // D = A(16x4 f32) * B(4x16 f32) + C(16x16 f32)
  return __builtin_amdgcn_wmma_f32_16x16x4_f32(false, a, false, b, (short)0, c,
                                               false, false);
}

__global__ __launch_bounds__(256) void fused_attn_ff(
    const float* __restrict__ e, const float* __restrict__ Wq,
    const float* __restrict__ Wk, const float* __restrict__ Wv,
    const float* __restrict__ Wo, const float* __restrict__ Wff,
    const float* __restrict__ bff, float* __restrict__ xbuf,
    float* __restrict__ part) {
  __shared__ float lds[LDS_FLOATS];
  float* Wb = lds;            // current weight, row-major [n][k] (=W as stored)
  float* Kb = lds + 16896;    // K  [key][d]
  float* Vb = lds + 33792;    // V^T [d][key]
  float* PW = lds + 50688;    // 8 waves x (16 x SW) scratch
  float* red = Kb;            // aliased after attention: [wave][2][128]

  const int tid  = threadIdx.x;
  const int lane = tid & 31;
  const int wv   = tid >> 5;   // 0..7
  const int lr   = lane & 15;  // M(A-row) / N(col)
  const int lh   = lane >> 4;  // k-pair / row-half selector
  const int wg = blockIdx.x;
  const int b = wg >> 7, i = wg & 127;
  float* pw = PW + wv * (16 * SW);

  // Warm L2 for the 5 shared weight matrices (global_prefetch_b8 path).
  {
    const float* wl[5] = {Wk, Wv, Wq, Wo, Wff};
#pragma unroll
    for (int s = 0; s < 5; ++s) {
      __builtin_prefetch(wl[s] + tid * 64, 0, 3);
      __builtin_prefetch(wl[s] + tid * 64 + 32, 0, 3);
    }
  }

  auto loadW = [&](const float* __restrict__ W) {
    for (int j = tid; j < 4096; j += 256) {
      int r = j >> 5, c4 = (j & 31) << 2;
      *(v4f*)(Wb + r * SW + c4) = *(const v4f*)(W + r * 128 + c4);
    }
  };

  // ---- A fragments for kv rows: e[b, i, 16w+lr, :] (L2-resident) ----
  v2f Aa[32];
  {
    const float* kvrow = e + ((((size_t)b * 128 + i) * 128) + (16 * wv + lr)) * 128;
#pragma unroll
    for (int ks = 0; ks < 32; ++ks) Aa[ks] = *(const v2f*)(kvrow + 4 * ks + 2 * lh);
  }

  // ---- K = kv @ Wk^T ----
  loadW(Wk);
  __syncthreads();
  for (int nt = 0; nt < 8; ++nt) {
    v8f acc = {0, 0, 0, 0, 0, 0, 0, 0};
#pragma unroll
    for (int ks = 0; ks < 32; ++ks) {
      v2f bf = *(const v2f*)(Wb + (16 * nt + lr) * SW + 4 * ks + 2 * lh);
      acc = wmma4(Aa[ks], bf, acc);
    }
#pragma unroll
    for (int r = 0; r < 8; ++r)
      Kb[(16 * wv + r + 8 * lh) * SW + 16 * nt + lr] = acc[r];
  }
  __syncthreads();

  // ---- V = kv @ Wv^T, stored transposed V^T[d][key] ----
  loadW(Wv);
  __syncthreads();
  for (int nt = 0; nt < 8; ++nt) {
    v8f acc = {0, 0, 0, 0, 0, 0, 0, 0};
#pragma unroll
    for (int ks = 0; ks < 32; ++ks) {
      v2f bf = *(const v2f*)(Wb + (16 * nt + lr) * SW + 4 * ks + 2 * lh);
      acc = wmma4(Aa[ks], bf, acc);
    }
#pragma unroll
    for (int r = 0; r < 8; ++r)
      Vb[(16 * nt + lr) * SW + 16 * wv + r + 8 * lh] = acc[r];
  }
  __syncthreads();

  // ---- Q = q_in @ Wq^T -> per-wave scratch ----
  {
    const float* qrow = e + ((((size_t)b * 128 + (16 * wv + lr)) * 128) + i) * 128;
#pragma unroll
    for (int ks = 0; ks < 32; ++ks) Aa[ks] = *(const v2f*)(qrow + 4 * ks + 2 * lh);
  }
  loadW(Wq);
  __syncthreads();
  for (int nt = 0; nt < 8; ++nt) {
    v8f acc = {0, 0, 0, 0, 0, 0, 0, 0};
#pragma unroll
    for (int ks = 0; ks < 32; ++ks) {
      v2f bf = *(const v2f*)(Wb + (16 * nt + lr) * SW + 4 * ks + 2 * lh);
      acc = wmma4(Aa[ks], bf, acc);
    }
#pragma unroll
    for (int r = 0; r < 8; ++r) pw[(r + 8 * lh) * SW + 16 * nt + lr] = acc[r];
  }
  __syncthreads();
  loadW(Wo);  // Wo consumed per-head inside attention
  __syncthreads();

  // ---- attention, fused with h_e += O_h @ Wo_h^T ----
  v8f HE[8];
#pragma unroll
  for (int nt = 0; nt < 8; ++nt) HE[nt] = (v8f){0, 0, 0, 0, 0, 0, 0, 0};

  for (int h = 0; h < 8; ++h) {
    v2f Qh[4];
#pragma unroll
    for (int ks = 0; ks < 4; ++ks)
      Qh[ks] = *(const v2f*)(pw + lr * SW + 16 * h + 4 * ks + 2 * lh);

    float rmx[8];
#pragma unroll
    for (int r = 0; r < 8; ++r) rmx[r] = -1e30f;
    for (int nt = 0; nt < 8; ++nt) {   // scores band: Q_h x K_h^T
      v8f acc = {0, 0, 0, 0, 0, 0, 0, 0};
#pragma unroll
      for (int ks = 0; ks < 4; ++ks) {
        v2f bf = *(const v2f*)(Kb + (16 * nt + lr) * SW + 16 * h + 4 * ks + 2 * lh);
        acc = wmma4(Qh[ks], bf, acc);
      }
#pragma unroll
      for (int r = 0; r < 8; ++r) {
        float s = acc[r] * 0.25f;  // 1/sqrt(16)
        pw[(r + 8 * lh) * SW + 16 * nt + lr] = s;
        rmx[r] = fmaxf(rmx[r], s);
      }
    }
#pragma unroll
    for (int m = 1; m < 16; m <<= 1)
#pragma unroll
      for (int r = 0; r < 8; ++r) rmx[r] = fmaxf(rmx[r], __shfl_xor(rmx[r], m, 32));

    float rsm[8];
#pragma unroll
    for (int r = 0; r < 8; ++r) rsm[r] = 0.f;
    for (int nt = 0; nt < 8; ++nt) {
#pragma unroll
      for (int r = 0; r < 8; ++r) {
        int idx = (r + 8 * lh) * SW + 16 * nt + lr;
        float v = __expf(pw[idx] - rmx[r]);
        pw[idx] = v;
        rsm[r] += v;
      }
    }
#pragma unroll
    for (int m = 1; m < 16; m <<= 1)
#pragma unroll
      for (int r = 0; r < 8; ++r) rsm[r] += __shfl_xor(rsm[r], m, 32);

    // O_h = exp(S) @ V_h (unnormalized), rows rescaled by 1/rowsum
    v8f accO = {0, 0, 0, 0, 0, 0, 0, 0};
    for (int ks = 0; ks < 32; ++ks) {
      v2f a = *(const v2f*)(pw + lr * SW + 4 * ks + 2 * lh);
      v2f bf = *(const v2f*)(Vb + (16 * h + lr) * SW + 4 * ks + 2 * lh);
      accO = wmma4(a, bf, accO);
    }
#pragma unroll
    for (int r = 0; r < 8; ++r)
      pw[(r + 8 * lh) * SW + 16 * h + lr] = accO[r] / rsm[r];

    // h_e += O_h @ Wo_h^T
    v2f ao[4];
#pragma unroll
    for (int ks = 0; ks < 4; ++ks)
      ao[ks] = *(const v2f*)(pw + lr * SW + 16 * h + 4 * ks + 2 * lh);
#pragma unroll
    for (int nt = 0; nt < 8; ++nt)
#pragma unroll
      for (int ks = 0; ks < 4; ++ks) {
        v2f bf = *(const v2f*)(Wb + (16 * nt + lr) * SW + 16 * h + 4 * ks + 2 * lh);
        HE[nt] = wmma4(ao[ks], bf, HE[nt]);
      }
  }

  // ---- stage h_e band, then FF + ReLU + residual + BN partials ----
#pragma unroll
  for (int nt = 0; nt < 8; ++nt)
#pragma unroll
    for (int r = 0; r < 8; ++r)
      pw[(r + 8 * lh) * SW + 16 * nt + lr] = HE[nt][r];
  __syncthreads();  // everyone done with Wo and Kb
  loadW(Wff);
  __syncthreads();

#pragma unroll
  for (int ks = 0; ks < 32; ++ks)
    Aa[ks] = *(const v2f*)(pw + lr * SW + 4 * ks + 2 * lh);

  for (int nt = 0; nt < 8; ++nt) {
    v8f acc = {0, 0, 0, 0, 0, 0, 0, 0};
#pragma unroll
    for (int ks = 0; ks < 32; ++ks) {
      v2f bf = *(const v2f*)(Wb + (16 * nt + lr) * SW + 4 * ks + 2 * lh);
      acc = wmma4(Aa[ks], bf, acc);
    }
    int col = 16 * nt + lr;
    float bias = bff[col];
    float s = 0.f, s2 = 0.f;
#pragma unroll
    for (int r = 0; r < 8; ++r) {
      int q = 16 * wv + r + 8 * lh;
      size_t g = ((((size_t)b * 128 + q) * 128) + i) * 128 + col;
      float v = fmaxf(acc[r] + bias, 0.f) + e[g];
      xbuf[g] = v;
      s += v;
      s2 += v * v;
    }
    s += __shfl_xor(s, 16, 32);
    s2 += __shfl_xor(s2, 16, 32);
    if (lh == 0) {
      red[(wv * 2 + 0) * 128 + col] = s;
      red[(wv * 2 + 1) * 128 + col] = s2;
    }
  }
  __syncthreads();
  {
    int which = tid >> 7, col = tid & 127;
    float p = 0.f;
#pragma unroll
    for (int w2 = 0; w2 < 8; ++w2) p += red[(w2 * 2 + which) * 128 + col];
    part[(size_t)wg * 256 + which * 128 + col] = p;
  }
}

__global__ void bn_finalize(const float* __restrict__ part,
                            const float* __restrict__ gamma,
                            const float* __restrict__ beta,
                            float* __restrict__ ss) {
  int d = threadIdx.x;
  if (d >= 128) return;
  float s = 0.f, s2 = 0.f;
#pragma unroll 8
  for (int wg = 0; wg < 512; ++wg) {
    s += part[wg * 256 + d];
    s2 += part[wg * 256 + 128 + d];
  }
  const float inv = 1.0f / 65536.0f;
  float mean = s * inv;
  float var = s2 * inv - mean * mean;
  float sc = gamma[d] * rsqrtf(var + 1e-5f);
  ss[d] = sc;
  ss[128 + d] = beta[d] - mean * sc;
}

// In-place: out holds pre-BN x (fully rewritten by fused_attn_ff each call).
__global__ __launch_bounds__(256) void bn_apply(float* __restrict__ out,
                                                const float* __restrict__ ss) {
  size_t idx = ((size_t)blockIdx.x * 256 + threadIdx.x) * 4;
  v4f x = *(const v4f*)(out + idx);
  int d = (int)(idx & 127);
  v4f sc = *(const v4f*)(ss + d);
  v4f sh = *(const v4f*)(ss + 128 + d);
  v4f r = x * sc + sh;
  *(v4f*)(out + idx) = r;
}

extern "C" void kernel_launch(void* const* d_in, const int* in_sizes, int n_in,
                              void* d_out, int out_size, void* d_ws,
                              size_t ws_size, hipStream_t stream) {
  const float* e = (const float*)d_in[0];
  const float* Wq = (const float*)d_in[1];
  const float* Wk = (const float*)d_in[2];
  const float* Wv = (const float*)d_in[3];
  const float* Wo = (const float*)d_in[4];
  const float* Wff = (const float*)d_in[5];
  const float* bff = (const float*)d_in[6];
  const float* gamma = (const float*)d_in[7];
  const float* beta = (const float*)d_in[8];

  float* xbuf = (float*)d_out;   // pre-BN x lives in d_out, normalized in place
  float* part = (float*)d_ws;    // 512 * 256 floats of BN partials
  float* ss = part + 131072;     // 256 floats: per-channel scale/shift

  fused_attn_ff<<<512, 256, 0, stream>>>(e, Wq, Wk, Wv, Wo, Wff, bff, xbuf, part);
  bn_finalize<<<1, 128, 0, stream>>>(part, gamma, beta, ss);
  bn_apply<<<8192, 256, 0, stream>>>(xbuf, ss);
}